// AverageSpanExtractor_13048110645573
// MI455X (gfx1250) — compile-verified
//
#include <hip/hip_runtime.h>

#ifndef __has_builtin
#define __has_builtin(x) 0
#endif

// Problem constants (from reference)
#define B_   8
#define S_   4096
#define D_   256
#define NS_  1024
#define W_   32
#define SPB  4          // spans per block

#if __has_builtin(__builtin_amdgcn_global_load_async_to_lds_b128) && \
    __has_builtin(__builtin_amdgcn_s_wait_asynccnt)
#define USE_ASYNC 1
#else
#define USE_ASYNC 0
#endif

// Builtin parameter types (from clang diagnostic):
//   arg0: __attribute__((vector_size(16))) int  addrspace(1)*   (global)
//   arg1: expected to be the LDS-side pointer, addrspace(3)
typedef int v4i __attribute__((vector_size(16)));
typedef __attribute__((address_space(1))) v4i* g_v4i_ptr;
typedef __attribute__((address_space(3))) v4i* l_v4i_ptr;

#if USE_ASYNC
// ---------------------------------------------------------------------------
// Async-to-LDS double-buffered pipeline.
// Each block: 256 threads (8 waves), 4 spans, 2 x 32KB LDS buffers.
// Per span each wave issues exactly 8 global_load_async_to_lds_b128 (32 lanes
// x 16B = 512B per instr; 8 waves x 8 instr = 32KB). Fixed instruction count
// => fixed-immediate s_wait_asynccnt pipelining is sound (in-order completion).
// ---------------------------------------------------------------------------
__global__ __launch_bounds__(256) void span_avg_async(const float* __restrict__ seq,
                                                      const int*   __restrict__ spans,
                                                      float*       __restrict__ out)
{
    __shared__ float buf[2][W_ * D_];   // 2 x 32KB
    const int tid       = threadIdx.x;
    const int base_span = blockIdx.x * SPB;

    // Issue this thread's 8 async 16B copies for span i into buffer `which`.
    auto issue = [&](int i, int which) {
        const int span = base_span + i;
        const int b    = span / NS_;
        const int st   = spans[2 * span + 0];
        const float* gsrc = seq + ((size_t)b * S_ + (size_t)st) * (size_t)D_;
#pragma unroll
        for (int k = 0; k < 8; ++k) {
            const int f4 = tid + k * 256;     // float4 index within 32KB tile
            __builtin_amdgcn_global_load_async_to_lds_b128(
                (g_v4i_ptr)(gsrc + (size_t)f4 * 4),
                (l_v4i_ptr)(&buf[which][f4 * 4]),
                /*offset=*/0, /*cpol=*/0);
        }
    };

    issue(0, 0);
#pragma unroll
    for (int i = 0; i < SPB; ++i) {
        if (i + 1 < SPB) {
            issue(i + 1, (i + 1) & 1);
            __builtin_amdgcn_s_wait_asynccnt(8);  // copy(i) done; copy(i+1) may fly
        } else {
            __builtin_amdgcn_s_wait_asynccnt(0);  // last copy fully done
        }
        __syncthreads();                          // all waves' slices visible

        const int span  = base_span + i;
        const int st    = spans[2 * span + 0];
        const int en    = spans[2 * span + 1];
        const int width = en - st;                // 1..32
        const float inv = 1.0f / (float)width;

        const float* lbuf = &buf[i & 1][0];
        float acc = 0.0f;
        for (int t = 0; t < width; ++t)
            acc += lbuf[t * D_ + tid];            // conflict-free DS reads
        out[(size_t)span * D_ + tid] = acc * inv;

        __syncthreads();                          // done reading before next overwrite
    }
}
#else
// ---------------------------------------------------------------------------
// Fallback: direct coalesced loads (compiles anywhere; no async builtins).
// ---------------------------------------------------------------------------
__global__ __launch_bounds__(256) void span_avg_direct(const float* __restrict__ seq,
                                                       const int*   __restrict__ spans,
                                                       float*       __restrict__ out)
{
    const int tid       = threadIdx.x;
    const int base_span = blockIdx.x * SPB;
#pragma unroll
    for (int i = 0; i < SPB; ++i) {
        const int span  = base_span + i;
        const int b     = span / NS_;
        const int st    = spans[2 * span + 0];
        const int en    = spans[2 * span + 1];
        const int width = en - st;
        const float* g  = seq + ((size_t)b * S_ + (size_t)st) * (size_t)D_ + tid;
        float acc = 0.0f;
        for (int t = 0; t < width; ++t)
            acc += g[(size_t)t * D_];
        out[(size_t)span * D_ + tid] = acc * (1.0f / (float)width);
    }
}
#endif

extern "C" void kernel_launch(void* const* d_in, const int* in_sizes, int n_in,
                              void* d_out, int out_size, void* d_ws, size_t ws_size,
                              hipStream_t stream)
{
    (void)in_sizes; (void)n_in; (void)d_ws; (void)ws_size; (void)out_size;
    const float* seq   = (const float*)d_in[0];
    const int*   spans = (const int*)d_in[1];
    float*       out   = (float*)d_out;

    const int n_spans = B_ * NS_;                 // 8192
    dim3 grid(n_spans / SPB);                     // 2048 blocks
    dim3 block(256);                              // 8 wave32s
#if USE_ASYNC
    span_avg_async<<<grid, block, 0, stream>>>(seq, spans, out);
#else
    span_avg_direct<<<grid, block, 0, stream>>>(seq, spans, out);
#endif
}